// Embedder_89017492177014
// MI455X (gfx1250) — compile-verified
//
#include <hip/hip_runtime.h>
#include <stdint.h>

// ---- CDNA5 TDM descriptor vector types (per CDNA5_HIP.md probe) ----
typedef uint32_t u32x4 __attribute__((ext_vector_type(4)));
typedef int32_t  i32x4 __attribute__((ext_vector_type(4)));
typedef int32_t  i32x8 __attribute__((ext_vector_type(8)));
typedef float    f32x4 __attribute__((ext_vector_type(4)));

#define D_MODEL 1024
#define VOCAB_N 32000
#define SEQ_N   2048
#define WPB     8            // waves (== tokens) per block, 256 threads
#define CHUNKS  (D_MODEL / 128)   // 8 float4s per lane per token

// out[t, :] = W[:, text[t]] + bias[:] + pe[t % SEQ, :]
//
// One wave per token. The strided column gather of W (stride = VOCAB*4 =
// 128KB between consecutive elements) is offloaded to the Tensor Data Mover:
// a 2D D# with tile_dim0=1 element, tile_dim1=1024 lines, dim0_stride=32000
// DMAs the whole column into this wave's LDS slice. While the DMA runs, the
// wave prefetches its bias and pe rows from global into registers, then waits
// on TENSORcnt and does the fused add + nontemporal store.
__global__ __launch_bounds__(WPB * 32) void
embed_pe_tdm_kernel(const int* __restrict__ text,
                    const float* __restrict__ W,
                    const float* __restrict__ bias,
                    const float* __restrict__ pe,
                    float* __restrict__ out)
{
    __shared__ float cols[WPB][D_MODEL];           // 32 KB / block

    const unsigned lane    = threadIdx.x & 31u;
    const unsigned wave    = threadIdx.x >> 5u;
    const unsigned tok_idx = blockIdx.x * WPB + wave;   // token index in [0, B*S)
    const unsigned s       = tok_idx & (SEQ_N - 1u);    // position in sequence

    // Keep the LDS allocation alive and give it a store so the compiler cannot
    // fold later loads of never-written shared memory to undef. The TDM engine
    // overwrites this at runtime.
    cols[wave][lane] = 0.0f;
    asm volatile("s_wait_dscnt 0" ::: "memory");   // order zero-store vs TDM write

    // Uniform-per-wave token id -> scalar for descriptor construction.
    int tok = __builtin_amdgcn_readfirstlane(text[tok_idx]);

    uint64_t gaddr   = (uint64_t)(uintptr_t)W + (uint64_t)(uint32_t)tok * 4ull;
    uint32_t lds_off = wave * (D_MODEL * 4u);      // byte offset of this wave's slice

    // ---- D# group 0 (128b): count=1 | lds_addr | global_addr[56:0] | type=2 ----
    u32x4 g0;
    g0[0] = 1u;                                    // count=1, is_restore=0, gather off
    g0[1] = lds_off;                               // lds_addr (bytes)
    g0[2] = (uint32_t)gaddr;                       // global_addr[31:0]
    g0[3] = ((uint32_t)(gaddr >> 32) & 0x01FFFFFFu) | (2u << 30); // addr[56:32] | type=2

    // ---- D# group 1 (256b) ----
    i32x8 g1;
    g1[0] = (int)(2u << 16);                       // wg_mask=0, data_size=2 (4B), no flags
    g1[1] = (int)((VOCAB_N & 0xFFFFu) << 16);      // barrier_addr=0 | tensor_dim0[15:0]
    g1[2] = (int)(((VOCAB_N >> 16) & 0xFFFFu) |    // tensor_dim0[31:16]
                  ((D_MODEL & 0xFFFFu) << 16));    // tensor_dim1[15:0]
    g1[3] = (int)(((D_MODEL >> 16) & 0xFFFFu) |    // tensor_dim1[31:16]
                  (1u << 16));                     // tile_dim0 = 1 element
    g1[4] = (int)D_MODEL;                          // tile_dim1 = 1024 lines, tile_dim2 = 0
    g1[5] = (int)VOCAB_N;                          // tensor_dim0_stride[31:0] = 32000
    g1[6] = 0;                                     // dim0_stride hi16 = 0, dim1_stride lo16 = 0
    g1[7] = 0;

    i32x4 gz = {0, 0, 0, 0};                       // groups 2/3 unused (2D tensor)

#if __clang_major__ >= 23
    i32x8 gz8 = {0, 0, 0, 0, 0, 0, 0, 0};
    __builtin_amdgcn_tensor_load_to_lds(g0, g1, gz, gz, gz8, 0);
#else
    __builtin_amdgcn_tensor_load_to_lds(g0, g1, gz, gz, 0);
#endif

    // Opaque zero index: blocks store-to-load forwarding of the dummy zeros.
    unsigned opq = 0;
    asm volatile("" : "+v"(opq));

    const f32x4* col4 = (const f32x4*)(&cols[wave][0]);
    const f32x4* b4   = (const f32x4*)bias;
    const f32x4* p4   = (const f32x4*)(pe + (size_t)s * D_MODEL);
    f32x4*       o4   = (f32x4*)(out + (size_t)tok_idx * D_MODEL);

    // Prefetch bias + pe into registers WHILE the TDM gather is in flight.
    // These are independent of LDS, so they overlap the DMA latency.
    f32x4 bb[CHUNKS], pp[CHUNKS];
#pragma unroll
    for (unsigned k = 0; k < CHUNKS; ++k) {
        unsigned idx = k * 32u + lane + opq;
        bb[k] = b4[idx];             // global_load_b128 (hot in L2)
        pp[k] = p4[idx];             // global_load_b128
    }

    __builtin_amdgcn_s_wait_tensorcnt(0);          // this wave's DMA complete
    asm volatile("" ::: "memory");

    // Fused add + write. Output is write-once and never re-read: use
    // nontemporal stores so the 32MB stream doesn't evict W/pe from L2.
#pragma unroll
    for (unsigned k = 0; k < CHUNKS; ++k) {
        unsigned idx = k * 32u + lane + opq;
        f32x4 c = col4[idx];         // ds_load_b128
        f32x4 r = c + bb[k] + pp[k];
        __builtin_nontemporal_store(r, &o4[idx]);  // global_store_b128 (NT)
    }
}

extern "C" void kernel_launch(void* const* d_in, const int* in_sizes, int n_in,
                              void* d_out, int out_size, void* d_ws, size_t ws_size,
                              hipStream_t stream) {
    (void)n_in; (void)out_size; (void)d_ws; (void)ws_size;
    const int*   text = (const int*)d_in[0];    // [B, S] int32
    const float* W    = (const float*)d_in[1];  // [D_MODEL, VOCAB] f32
    const float* b    = (const float*)d_in[2];  // [D_MODEL] f32
    const float* pe   = (const float*)d_in[3];  // [MAX_LEN, D_MODEL] f32
    float*       out  = (float*)d_out;          // [B, S, D_MODEL] f32

    const int n_tokens = in_sizes[0];           // B*S = 8192
    const int blocks   = n_tokens / WPB;        // 1024 blocks, 256 threads each

    hipLaunchKernelGGL(embed_pe_tdm_kernel, dim3(blocks), dim3(WPB * 32), 0, stream,
                       text, W, b, pe, out);
}